// FullTripletLoss_42631845380801
// MI455X (gfx1250) — compile-verified
//
#include <hip/hip_runtime.h>
#include <hip/hip_bf16.h>

typedef __attribute__((ext_vector_type(2))) float v2f;
typedef __attribute__((ext_vector_type(8))) float v8f;

#define NPTS 256
#define CDIM 512
#define MDIM 64
#define KPOS 8
#define TMARGIN 0.2f

#define TS   32          // n,k tile edge per block
#define MB   16          // m-slices per block
#define KC   8           // c-depth staged per chunk
#define RSTR 130         // padded row stride (4 pairs * 32 + 2) -> conflict-free b64 reads
#define LSZ  (TS * RSTR) // 4160 floats per matrix

// ---------------------------------------------------------------------------
// Kernel 1: batched Gram G[m][n][k] = sum_c feature[n,c,m]*feature[k,c,m]
// Block: 32x32 (n,k) tile x 16 m-slices. 8 waves; wave owns 2 m x (2x2) 16x16
// subtiles = 8 v8f accumulators. LDS pair-interleaved [row][cpair][m][2] so a
// WMMA operand is one aligned ds_load_b64 straight into a v2f.
// Register-held prefetch of the next c-chunk overlaps global latency w/ WMMA.
// ---------------------------------------------------------------------------
__global__ __launch_bounds__(256) void gram_kernel(const float* __restrict__ feat,
                                                   float* __restrict__ G) {
  const int k0 = blockIdx.x * TS;
  const int n0 = blockIdx.y * TS;
  const int m0 = blockIdx.z * MB;
  const int tid  = threadIdx.x;
  const int wave = tid >> 5;
  const int lane = tid & 31;
  const int half = lane >> 4;   // selects K=0,1 vs K=2,3 within a 4-step
  const int lr   = lane & 15;   // A row / B col within a 16x16 fragment

  __shared__ __align__(16) float ldsA[LSZ];
  __shared__ __align__(16) float ldsB[LSZ];

  v8f acc[2][2][2] = {};        // [mm][ti][tj]

  // staging: 256 (row,c) pairs per matrix, 4 m per float4
  const int tq = tid & 3;       // m-quad: m = m0 + 4*tq .. +3
  const int gp = tid >> 2;      // 0..63 -> 4 (row,c) pairs each
  const size_t mbase = (size_t)(m0 + tq * 4);

  float4 ra[4], rb[4];

  // ---- prologue: stage chunk 0 ----
#pragma unroll
  for (int r = 0; r < 4; ++r) {
    const int pair = gp * 4 + r;           // 0..255
    const int row  = pair >> 3;
    const int j    = pair & 7;
    const size_t coff = (size_t)j * MDIM + mbase;
    ra[r] = *reinterpret_cast<const float4*>(
        &feat[(size_t)(n0 + row) * (CDIM * MDIM) + coff]);
    rb[r] = *reinterpret_cast<const float4*>(
        &feat[(size_t)(k0 + row) * (CDIM * MDIM) + coff]);
  }
#pragma unroll
  for (int r = 0; r < 4; ++r) {
    const int pair = gp * 4 + r;
    const int row  = pair >> 3;
    const int j    = pair & 7;
    const int base = row * RSTR + (j >> 1) * 32 + (j & 1);
    ldsA[base + (tq * 4 + 0) * 2] = ra[r].x;
    ldsA[base + (tq * 4 + 1) * 2] = ra[r].y;
    ldsA[base + (tq * 4 + 2) * 2] = ra[r].z;
    ldsA[base + (tq * 4 + 3) * 2] = ra[r].w;
    ldsB[base + (tq * 4 + 0) * 2] = rb[r].x;
    ldsB[base + (tq * 4 + 1) * 2] = rb[r].y;
    ldsB[base + (tq * 4 + 2) * 2] = rb[r].z;
    ldsB[base + (tq * 4 + 3) * 2] = rb[r].w;
  }
  __syncthreads();

  const int NCHUNK = CDIM / KC;  // 64
  for (int it = 0; it < NCHUNK; ++it) {
    // ---- prefetch next chunk into registers (overlaps WMMA below) ----
    if (it + 1 < NCHUNK) {
      const int c0 = (it + 1) * KC;
#pragma unroll
      for (int r = 0; r < 4; ++r) {
        const int pair = gp * 4 + r;
        const int row  = pair >> 3;
        const int j    = pair & 7;
        const size_t coff = (size_t)(c0 + j) * MDIM + mbase;
        ra[r] = *reinterpret_cast<const float4*>(
            &feat[(size_t)(n0 + row) * (CDIM * MDIM) + coff]);
        rb[r] = *reinterpret_cast<const float4*>(
            &feat[(size_t)(k0 + row) * (CDIM * MDIM) + coff]);
      }
    }

    // ---- compute on current LDS chunk: 2 m x 2 k-steps x 2x2 tiles ----
#pragma unroll
    for (int mm = 0; mm < 2; ++mm) {
      const int ml = wave * 2 + mm;
#pragma unroll
      for (int jj2 = 0; jj2 < 2; ++jj2) {     // k-step pairs: c = 4*jj2(+2*half)
        const int jp = jj2 * 2 + half;
        v2f av[2], bv[2];
#pragma unroll
        for (int ti = 0; ti < 2; ++ti)
          av[ti] = *reinterpret_cast<const v2f*>(
              &ldsA[(ti * 16 + lr) * RSTR + jp * 32 + ml * 2]);
#pragma unroll
        for (int tj = 0; tj < 2; ++tj)
          bv[tj] = *reinterpret_cast<const v2f*>(
              &ldsB[(tj * 16 + lr) * RSTR + jp * 32 + ml * 2]);
#pragma unroll
        for (int ti = 0; ti < 2; ++ti)
#pragma unroll
          for (int tj = 0; tj < 2; ++tj)
            acc[mm][ti][tj] = __builtin_amdgcn_wmma_f32_16x16x4_f32(
                false, av[ti], false, bv[tj], (short)0, acc[mm][ti][tj],
                false, false);
      }
    }
    __syncthreads();   // all reads of current chunk done

    // ---- commit prefetched registers to LDS ----
    if (it + 1 < NCHUNK) {
#pragma unroll
      for (int r = 0; r < 4; ++r) {
        const int pair = gp * 4 + r;
        const int row  = pair >> 3;
        const int j    = pair & 7;
        const int base = row * RSTR + (j >> 1) * 32 + (j & 1);
        ldsA[base + (tq * 4 + 0) * 2] = ra[r].x;
        ldsA[base + (tq * 4 + 1) * 2] = ra[r].y;
        ldsA[base + (tq * 4 + 2) * 2] = ra[r].z;
        ldsA[base + (tq * 4 + 3) * 2] = ra[r].w;
        ldsB[base + (tq * 4 + 0) * 2] = rb[r].x;
        ldsB[base + (tq * 4 + 1) * 2] = rb[r].y;
        ldsB[base + (tq * 4 + 2) * 2] = rb[r].z;
        ldsB[base + (tq * 4 + 3) * 2] = rb[r].w;
      }
      __syncthreads();
    }
  }

  // ---- writeback: VGPR r holds rows r (lanes 0-15) / r+8 (lanes 16-31) ----
#pragma unroll
  for (int mm = 0; mm < 2; ++mm) {
    const int m = m0 + wave * 2 + mm;
#pragma unroll
    for (int ti = 0; ti < 2; ++ti)
#pragma unroll
      for (int tj = 0; tj < 2; ++tj)
#pragma unroll
        for (int r = 0; r < 8; ++r) {
          const int row = ti * 16 + r + half * 8;
          const int col = tj * 16 + lr;
          G[(size_t)m * NPTS * NPTS + (size_t)(n0 + row) * NPTS + (k0 + col)] =
              acc[mm][ti][tj][r];
        }
  }
}

// ---------------------------------------------------------------------------
// Kernel 2: per (m, anchor n) triplet reduction. Thread k owns column k.
// ---------------------------------------------------------------------------
__global__ __launch_bounds__(256) void triplet_kernel(const float* __restrict__ G,
                                                      const int* __restrict__ label,
                                                      float* __restrict__ perM_sum,
                                                      float* __restrict__ perM_cnt) {
  const int n = blockIdx.x;
  const int m = blockIdx.y;
  const int k = threadIdx.x;
  const float* Gm = G + (size_t)m * NPTS * NPTS;

  const float gnn = Gm[(size_t)n * NPTS + n];
  const float gkk = Gm[(size_t)k * NPTS + k];
  const float gnk = Gm[(size_t)n * NPTS + k];
  const float z = gnn + gkk - 2.0f * gnk;
  const float d = (z > 0.0f) ? sqrtf(z) : 0.0f;   // _safe_sqrt

  __shared__ float dap[KPOS];
  const bool pos = (label[k] == label[n]);
  if (pos) dap[k >> 5] = d;                        // positives: k = n%32 + 32*(k>>5)
  __syncthreads();

  float s = 0.0f, c = 0.0f;
  if (!pos) {
#pragma unroll
    for (int p = 0; p < KPOS; ++p) {
      const float t = TMARGIN + dap[p] - d;
      if (t > 0.0f) { s += t; c += 1.0f; }
    }
  }

  __shared__ float rs[256], rc[256];
  rs[k] = s; rc[k] = c;
  __syncthreads();
  for (int off = 128; off > 0; off >>= 1) {
    if (k < off) { rs[k] += rs[k + off]; rc[k] += rc[k + off]; }
    __syncthreads();
  }
  if (k == 0) {
    atomicAdd(&perM_sum[m], rs[0]);
    atomicAdd(&perM_cnt[m], rc[0]);
  }
}

__global__ void zero_kernel(float* __restrict__ p, int n) {
  const int i = blockIdx.x * blockDim.x + threadIdx.x;
  if (i < n) p[i] = 0.0f;
}

__global__ __launch_bounds__(64) void finalize_kernel(const float* __restrict__ perM_sum,
                                                      const float* __restrict__ perM_cnt,
                                                      float* __restrict__ out) {
  __shared__ float sm[64], sc[64];
  const int t = threadIdx.x;
  const float s = perM_sum[t];
  const float c = perM_cnt[t];
  sm[t] = (c > 0.0f) ? (s / c) : 0.0f;
  sc[t] = c;
  __syncthreads();
  for (int off = 32; off > 0; off >>= 1) {
    if (t < off) { sm[t] += sm[t + off]; sc[t] += sc[t + off]; }
    __syncthreads();
  }
  if (t == 0) {
    out[0] = sm[0] / (float)MDIM;
    out[1] = (sc[0] / (float)MDIM) / (float)(NPTS * KPOS * (NPTS - KPOS));
  }
}

extern "C" void kernel_launch(void* const* d_in, const int* in_sizes, int n_in,
                              void* d_out, int out_size, void* d_ws, size_t ws_size,
                              hipStream_t stream) {
  const float* feat  = (const float*)d_in[0];
  const int*   label = (const int*)d_in[1];

  // Workspace: G (64*256*256 f32 = 16 MB), then perM_sum[64], perM_cnt[64].
  float* G        = (float*)d_ws;
  float* perM     = G + (size_t)MDIM * NPTS * NPTS;
  float* perM_sum = perM;
  float* perM_cnt = perM + MDIM;

  zero_kernel<<<1, 128, 0, stream>>>(perM, 2 * MDIM);

  dim3 g1(NPTS / TS, NPTS / TS, MDIM / MB);   // 8 x 8 x 4
  gram_kernel<<<g1, 256, 0, stream>>>(feat, G);

  dim3 g2(NPTS, MDIM);                        // (anchor n, slice m)
  triplet_kernel<<<g2, 256, 0, stream>>>(G, label, perM_sum, perM_cnt);

  finalize_kernel<<<1, 64, 0, stream>>>(perM_sum, perM_cnt, (float*)d_out);
}